// QuantumSelectiveSSM_74431783239899
// MI455X (gfx1250) — compile-verified
//
#include <hip/hip_runtime.h>
#include <hip/hip_bf16.h>

// ---------------------------------------------------------------------------
// 12-qubit statevector simulator, batch=256, one workgroup per batch element.
// State (4096 complex) lives in LDS as split re/im float arrays (32 KB).
// Single-qubit gates are fused per layer:   G_i = RZ*RY,  H_i = RZ(c)*RX*RY(b)
// and groups of 4 qubits are tensor-producted into 16x16 complex matrices that
// are applied with V_WMMA_F32_16X16X4_F32 (4 k-chunks x 4 real-GEMM combos).
// CRX chain + expectations are scalar LDS passes.
// ---------------------------------------------------------------------------

typedef __attribute__((ext_vector_type(2))) float v2f;
typedef __attribute__((ext_vector_type(8))) float v8f;

#define NQ    12
#define DIM   4096
#define NTHR  256

__device__ __forceinline__ v8f wmma4(v2f a, v2f b, v8f c) {
  // D = A(16x4) x B(4x16) + C(16x16), f32. 8-arg form (neg flags unused for f32).
  return __builtin_amdgcn_wmma_f32_16x16x4_f32(
      false, a, false, b, (short)0, c, false, false);
}

// address of element 'a' (0..15) of 16-vector 'v' (0..255) for qubit group g
//   g=0: qubits 0-3  -> flat bits 11..8 (stride 256)
//   g=1: qubits 4-7  -> flat bits  7..4 (stride 16)
//   g=2: qubits 8-11 -> flat bits  3..0 (stride 1)
__device__ __forceinline__ int grpAddr(int g, int v, int a) {
  if (g == 0) return v + (a << 8);
  if (g == 1) return ((v >> 4) << 8) + (v & 15) + (a << 4);
  return (v << 4) + a;
}

__device__ float blockReduce(float val, float* scratch, int tid) {
  #pragma unroll
  for (int off = 16; off > 0; off >>= 1)
    val += __shfl_down(val, off, 32);
  __syncthreads();                 // scratch free from previous use
  if ((tid & 31) == 0) scratch[tid >> 5] = val;
  __syncthreads();
  return scratch[0] + scratch[1] + scratch[2] + scratch[3]
       + scratch[4] + scratch[5] + scratch[6] + scratch[7];
}

// Apply 16x16 complex matrix (sMre/sMim, row-major M[a*16+b]) to group g.
// One wave handles 2 tiles of 16 vectors; 8 waves cover all 256 vectors.
__device__ void applyGroup(int g, float* sRe, float* sIm,
                           const float* sMre, const float* sMim, int tid) {
  const int lane = tid & 31;
  const int wave = tid >> 5;
  const int n    = lane & 15;     // column (N) for B/C/D fragments
  const int half = lane >> 4;     // lane-half selects K (A/B) or M-offset (D)
  const int m    = lane & 15;     // row (M) for A fragment

  // A fragments of U: k = ch*4 + 2*half + j  (ISA 16x4 f32 A layout)
  v2f Are[4], Aim[4], AimN[4];
  #pragma unroll
  for (int ch = 0; ch < 4; ++ch) {
    #pragma unroll
    for (int j = 0; j < 2; ++j) {
      int k = ch * 4 + half * 2 + j;
      float ur = sMre[m * 16 + k];
      float ui = sMim[m * 16 + k];
      Are[ch][j]  = ur;
      Aim[ch][j]  = ui;
      AimN[ch][j] = -ui;          // f32 WMMA has no A-neg modifier: pre-negate
    }
  }

  #pragma unroll
  for (int tt = 0; tt < 2; ++tt) {
    const int tile  = wave * 2 + tt;
    const int v     = tile * 16 + n;

    // B fragments of X: row k = ch*4 + 2*half + j (mirror of A's K mapping)
    v2f Bre[4], Bim[4];
    #pragma unroll
    for (int ch = 0; ch < 4; ++ch) {
      #pragma unroll
      for (int j = 0; j < 2; ++j) {
        int a = ch * 4 + half * 2 + j;
        int addr = grpAddr(g, v, a);
        Bre[ch][j] = sRe[addr];
        Bim[ch][j] = sIm[addr];
      }
    }

    v8f accRe = {0.f,0.f,0.f,0.f,0.f,0.f,0.f,0.f};
    v8f accIm = {0.f,0.f,0.f,0.f,0.f,0.f,0.f,0.f};
    #pragma unroll
    for (int ch = 0; ch < 4; ++ch) {
      accRe = wmma4(Are[ch],  Bre[ch], accRe);   // +Ur*Xr
      accRe = wmma4(AimN[ch], Bim[ch], accRe);   // -Ui*Xi
      accIm = wmma4(Are[ch],  Bim[ch], accIm);   // +Ur*Xi
      accIm = wmma4(Aim[ch],  Bre[ch], accIm);   // +Ui*Xr
    }

    // D layout: VGPR r -> row a = r + 8*half, col n
    #pragma unroll
    for (int r = 0; r < 8; ++r) {
      int a = r + half * 8;
      int addr = grpAddr(g, v, a);
      sRe[addr] = accRe[r];
      sIm[addr] = accIm[r];
    }
  }
}

__global__ __launch_bounds__(NTHR)
void QuantumSelectiveSSM_74431783239899_kernel(const float* __restrict__ base,
                                               const float* __restrict__ qp,
                                               const float* __restrict__ bp,
                                               const float* __restrict__ cp,
                                               float* __restrict__ out) {
  __shared__ float sRe[DIM];
  __shared__ float sIm[DIM];
  __shared__ float sMre[256];
  __shared__ float sMim[256];
  __shared__ float sG[12 * 8];   // per-qubit fused 2x2 complex gate
  __shared__ float sRed[8];

  const int tid = threadIdx.x;
  const int bid = blockIdx.x;

  // ---- load + normalize -------------------------------------------------
  float acc = 0.f;
  for (int i = tid; i < DIM; i += NTHR) {
    float v = base[(size_t)bid * DIM + i];
    sRe[i] = v;
    sIm[i] = 0.f;
    acc += v * v;
  }
  float nrm2 = blockReduce(acc, sRed, tid);
  float inv = rsqrtf(nrm2);
  for (int i = tid; i < DIM; i += NTHR) sRe[i] *= inv;
  __syncthreads();

  // ---- layers -----------------------------------------------------------
  for (int layer = 0; layer < 4; ++layer) {
    const int pbase = layer * 47;   // 24 (ry/rz) + 11 (crx) + 12 (rx)

    // Phase A: fused G_i = RZ(p1)*RY(p0) per qubit
    if (tid < NQ) {
      float s, c, sf, cf;
      __sincosf(qp[pbase + 2 * tid]     * 0.5f, &s,  &c);
      __sincosf(qp[pbase + 2 * tid + 1] * 0.5f, &sf, &cf);
      float* Gq = &sG[tid * 8];
      Gq[0] =  c * cf; Gq[1] = -c * sf;   // G00 = c*e^{-i phi/2}
      Gq[2] = -s * cf; Gq[3] =  s * sf;   // G01 = -s*e^{-i phi/2}
      Gq[4] =  s * cf; Gq[5] =  s * sf;   // G10 =  s*e^{+i phi/2}
      Gq[6] =  c * cf; Gq[7] =  c * sf;   // G11 =  c*e^{+i phi/2}
    }
    __syncthreads();

    for (int g = 0; g < 3; ++g) {
      // build M = G_{4g} (x) G_{4g+1} (x) G_{4g+2} (x) G_{4g+3}
      {
        int a = tid >> 4, b = tid & 15;
        float pr = 1.f, pi = 0.f;
        #pragma unroll
        for (int j = 0; j < 4; ++j) {
          int abit = (a >> (3 - j)) & 1;
          int bbit = (b >> (3 - j)) & 1;
          const float* Gq = &sG[(4 * g + j) * 8];
          float gr = Gq[(abit * 2 + bbit) * 2];
          float gi = Gq[(abit * 2 + bbit) * 2 + 1];
          float nr = pr * gr - pi * gi;
          float ni = pr * gi + pi * gr;
          pr = nr; pi = ni;
        }
        sMre[a * 16 + b] = pr;
        sMim[a * 16 + b] = pi;
      }
      __syncthreads();
      applyGroup(g, sRe, sIm, sMre, sMim, tid);
      __syncthreads();
    }

    // Phase B: CRX chain, control qubit i (bit 11-i), target i+1 (bit 10-i)
    for (int i = 0; i < 11; ++i) {
      float s, c;
      __sincosf(qp[pbase + 24 + i] * 0.5f, &s, &c);
      const int tb = 10 - i;
      #pragma unroll
      for (int kk = 0; kk < 4; ++kk) {
        int k  = tid + kk * NTHR;          // 1024 pairs
        int lo = k & ((1 << tb) - 1);
        int hi = k >> tb;
        int i0 = (hi << (tb + 2)) | (1 << (tb + 1)) | lo;  // ctrl=1, tgt=0
        int i1 = i0 | (1 << tb);
        float r0 = sRe[i0], m0 = sIm[i0];
        float r1 = sRe[i1], m1 = sIm[i1];
        sRe[i0] = c * r0 + s * m1;
        sIm[i0] = c * m0 - s * r1;
        sRe[i1] = c * r1 + s * m0;
        sIm[i1] = c * m1 - s * r0;
      }
      __syncthreads();
    }

    // Phase C: fused H_i = RZ(c_i)*RX(p)*RY(b_i)
    if (tid < NQ) {
      float sb, cb, st, ct, sc, cc;
      __sincosf(bp[tid]               * 0.5f, &sb, &cb);
      __sincosf(qp[pbase + 35 + tid]  * 0.5f, &st, &ct);
      __sincosf(cp[tid]               * 0.5f, &sc, &cc);
      // P = RX*RY
      float P00r =  ct * cb, P00i = -st * sb;
      float P01r = -ct * sb, P01i = -st * cb;
      float P10r =  ct * sb, P10i = -st * cb;
      float P11r =  ct * cb, P11i =  st * sb;
      float* Gq = &sG[tid * 8];
      // row0 *= e^{-i c/2}, row1 *= e^{+i c/2}
      Gq[0] = cc * P00r + sc * P00i; Gq[1] = cc * P00i - sc * P00r;
      Gq[2] = cc * P01r + sc * P01i; Gq[3] = cc * P01i - sc * P01r;
      Gq[4] = cc * P10r - sc * P10i; Gq[5] = cc * P10i + sc * P10r;
      Gq[6] = cc * P11r - sc * P11i; Gq[7] = cc * P11i + sc * P11r;
    }
    __syncthreads();

    for (int g = 0; g < 3; ++g) {
      {
        int a = tid >> 4, b = tid & 15;
        float pr = 1.f, pi = 0.f;
        #pragma unroll
        for (int j = 0; j < 4; ++j) {
          int abit = (a >> (3 - j)) & 1;
          int bbit = (b >> (3 - j)) & 1;
          const float* Gq = &sG[(4 * g + j) * 8];
          float gr = Gq[(abit * 2 + bbit) * 2];
          float gi = Gq[(abit * 2 + bbit) * 2 + 1];
          float nr = pr * gr - pi * gi;
          float ni = pr * gi + pi * gr;
          pr = nr; pi = ni;
        }
        sMre[a * 16 + b] = pr;
        sMim[a * 16 + b] = pi;
      }
      __syncthreads();
      applyGroup(g, sRe, sIm, sMre, sMim, tid);
      __syncthreads();
    }
  }

  // ---- expectations: per qubit w: 2Re<z01>, 2Im<z01>, <|a0|^2-|a1|^2> ----
  for (int w = 0; w < NQ; ++w) {
    const int bpos = 11 - w;
    const int S = 1 << bpos;
    float zr = 0.f, zi = 0.f, dd = 0.f;
    #pragma unroll
    for (int kk = 0; kk < 8; ++kk) {
      int k  = tid + kk * NTHR;            // 2048 pairs
      int lo = k & (S - 1);
      int hi = k >> bpos;
      int i0 = (hi << (bpos + 1)) | lo;
      int i1 = i0 | S;
      float r0 = sRe[i0], m0 = sIm[i0];
      float r1 = sRe[i1], m1 = sIm[i1];
      zr += r0 * r1 + m0 * m1;             // Re(conj(a0)*a1)
      zi += r0 * m1 - m0 * r1;             // Im(conj(a0)*a1)
      dd += (r0 * r0 + m0 * m0) - (r1 * r1 + m1 * m1);
    }
    zr = blockReduce(zr, sRed, tid);
    zi = blockReduce(zi, sRed, tid);
    dd = blockReduce(dd, sRed, tid);
    if (tid == 0) {
      out[bid * 36 + 3 * w + 0] = 2.f * zr;
      out[bid * 36 + 3 * w + 1] = 2.f * zi;
      out[bid * 36 + 3 * w + 2] = dd;
    }
  }
}

extern "C" void kernel_launch(void* const* d_in, const int* in_sizes, int n_in,
                              void* d_out, int out_size, void* d_ws, size_t ws_size,
                              hipStream_t stream) {
  const float* base = (const float*)d_in[0];   // (BATCH, 4096) f32
  const float* qp   = (const float*)d_in[1];   // 192 f32
  const float* bpar = (const float*)d_in[2];   // 12 f32
  const float* cpar = (const float*)d_in[3];   // 12 f32
  float* out = (float*)d_out;                  // (BATCH, 36) f32
  const int batch = in_sizes[0] / DIM;
  QuantumSelectiveSSM_74431783239899_kernel<<<batch, NTHR, 0, stream>>>(
      base, qp, bpar, cpar, out);
}